// DistanceLoss_16484084483043
// MI455X (gfx1250) — compile-verified
//
#include <hip/hip_runtime.h>

typedef float fv4 __attribute__((ext_vector_type(4)));
typedef float v2f __attribute__((ext_vector_type(2)));
typedef float v8f __attribute__((ext_vector_type(8)));

#define T_  64
#define H_  32
#define W_  8
#define B_  2048
#define B4_ (B_ / 4)     // 512 float4's along B
#define HS_ 2            // h-chunks for extra parallelism
#define HC_ (H_ / HS_)   // 16 h per chunk

// ---------------------------------------------------------------------------
// Kernel 1: pure streaming pass over the two big [T,H,W,B] tensors.
// For one (t, h-chunk), each thread owns 4 consecutive b's (float4 loads,
// 512B per wave per instruction). Computes:
//   x_h = ofd[t,h,b] + sum_w inner[t,h,w,b]
//   y_h = ifd[t,h,b] + sum_w outer[t,h,w,b]
// and tracks min_h(x), max_h(y) over the chunk. Scales are applied later
// (valid since the scale factors are >= 0 and h-independent).
// ---------------------------------------------------------------------------
__global__ __launch_bounds__(256)
void k_partial(const float* __restrict__ inner,   // [T,H,W,B]
               const float* __restrict__ outer,   // [T,H,W,B]
               const float* __restrict__ ofd,     // [T,H,B]
               const float* __restrict__ ifd,     // [T,H,B]
               float* __restrict__ minp,          // [HS,T,B]
               float* __restrict__ maxp)          // [HS,T,B]
{
    const int b4 = blockIdx.x * blockDim.x + threadIdx.x;   // 0..B4_-1
    const int t  = blockIdx.y;
    const int hz = blockIdx.z;

    const fv4* in4 = (const fv4*)inner;
    const fv4* ou4 = (const fv4*)outer;
    const fv4* of4 = (const fv4*)ofd;
    const fv4* if4 = (const fv4*)ifd;

    fv4 xmin = {3.4e38f, 3.4e38f, 3.4e38f, 3.4e38f};
    fv4 ymax = {-3.4e38f, -3.4e38f, -3.4e38f, -3.4e38f};

    const int h0 = hz * HC_;
    for (int h = h0; h < h0 + HC_; ++h) {
        const int th = t * H_ + h;
        fv4 iw = {0.f, 0.f, 0.f, 0.f};
        fv4 ow = {0.f, 0.f, 0.f, 0.f};
#pragma unroll
        for (int w = 0; w < W_; ++w) {
            const int idx = (th * W_ + w) * B4_ + b4;
            iw += __builtin_nontemporal_load(&in4[idx]);   // streamed once -> NT
            ow += __builtin_nontemporal_load(&ou4[idx]);
        }
        fv4 x = __builtin_nontemporal_load(&of4[th * B4_ + b4]) + iw;
        fv4 y = __builtin_nontemporal_load(&if4[th * B4_ + b4]) + ow;
        xmin = __builtin_elementwise_min(xmin, x);
        ymax = __builtin_elementwise_max(ymax, y);
    }

    ((fv4*)minp)[(hz * T_ + t) * B4_ + b4] = xmin;
    ((fv4*)maxp)[(hz * T_ + t) * B4_ + b4] = ymax;
}

// ---------------------------------------------------------------------------
// Kernel 2: combine the HS h-chunks, apply the belongs/containment scales,
// sum over T. One thread per b. Reads ~2 MB total -> negligible.
// ---------------------------------------------------------------------------
__global__ __launch_bounds__(256)
void k_reduce_t(const float* __restrict__ minp,           // [HS,T,B]
                const float* __restrict__ maxp,           // [HS,T,B]
                const float* __restrict__ cont,           // [T,B]
                const unsigned char* __restrict__ target, // [B,T] bool
                float* __restrict__ colsum)               // [B]
{
    const int b = blockIdx.x * blockDim.x + threadIdx.x;  // 0..B_-1
    float acc = 0.f;
    for (int t = 0; t < T_; ++t) {
        float x = fminf(minp[t * B_ + b], minp[(T_ + t) * B_ + b]);
        float y = fmaxf(maxp[t * B_ + b], maxp[(T_ + t) * B_ + b]);
        float bel = target[b * T_ + t] ? 1.f : 0.f;
        float c   = cont[t * B_ + b];
        acc += bel * (1.f - c) * x + (1.f - bel) * c * y;
    }
    colsum[b] = acc;
}

// ---------------------------------------------------------------------------
// Kernel 3: single wave32. Each lane accumulates 64 column sums, then the
// 32 lane-partials are reduced with V_WMMA_F32_16X16X4_F32:
//   A[m,0] = s_m, A[m,2] = s_{m+16} (other K slots 0), B = ones
//   => D[m,n] = s_m + s_{m+16} for all n.
// Summing the 8 D VGPRs per lane gives half-sums in lanes 0-15 / 16-31;
// one shfl_xor(16) completes the reduction. All math stays in f32 (RNE).
// ---------------------------------------------------------------------------
__global__ __launch_bounds__(32)
void k_final(const float* __restrict__ colsum, float* __restrict__ out)
{
    const int lane = threadIdx.x;
    float s = 0.f;
    for (int i = 0; i < B_ / 32; ++i) s += colsum[lane + 32 * i];

    v2f a;  a.x = s;   a.y = 0.f;   // A 16x4: K1/K3 slots zero
    v2f bo; bo.x = 1.f; bo.y = 1.f; // B 4x16: all ones (layout-independent)
    v8f c = {};
    c = __builtin_amdgcn_wmma_f32_16x16x4_f32(
        /*neg_a=*/false, a, /*neg_b=*/false, bo,
        /*c_mod=*/(short)0, c, /*reuse_a=*/false, /*reuse_b=*/false);

    float v = c[0] + c[1] + c[2] + c[3] + c[4] + c[5] + c[6] + c[7];
    v += __shfl_xor(v, 16, 32);
    if (lane == 0) out[0] = v * (1.0f / (float)B_);
}

// ---------------------------------------------------------------------------
extern "C" void kernel_launch(void* const* d_in, const int* in_sizes, int n_in,
                              void* d_out, int out_size, void* d_ws, size_t ws_size,
                              hipStream_t stream)
{
    const float*         inner  = (const float*)d_in[0];
    const float*         outer  = (const float*)d_in[1];
    const float*         ofd    = (const float*)d_in[2];
    const float*         ifd    = (const float*)d_in[3];
    const float*         cont   = (const float*)d_in[4];
    const unsigned char* target = (const unsigned char*)d_in[5];

    float* wsf    = (float*)d_ws;
    float* minp   = wsf;                              // HS*T*B floats
    float* maxp   = wsf + (size_t)HS_ * T_ * B_;      // HS*T*B floats
    float* colsum = wsf + (size_t)2 * HS_ * T_ * B_;  // B floats

    dim3 g1(B4_ / 256, T_, HS_);  // (2, 64, 2) = 256 blocks x 8 waves
    k_partial<<<g1, 256, 0, stream>>>(inner, outer, ofd, ifd, minp, maxp);
    k_reduce_t<<<B_ / 256, 256, 0, stream>>>(minp, maxp, cont, target, colsum);
    k_final<<<1, 32, 0, stream>>>(colsum, (float*)d_out);
}